// DeepSeekSparseAttention_23570780520572
// MI455X (gfx1250) — compile-verified
//
#include <hip/hip_runtime.h>
#include <hip/hip_bf16.h>

// ---------------- dims ----------------
#define B_ 2
#define N_ 4096
#define D_ 384
#define NHEADS 8
#define KDIM 48
#define VDIM 48
#define TOPK 128
#define IHEADS 2
#define IDIM 32
#define CHUNK_Q 512
#define SCALE_F 0.14433756729740643f  // 48^-0.5

typedef float v2f __attribute__((ext_vector_type(2)));
typedef float v8f __attribute__((ext_vector_type(8)));

#if __has_builtin(__builtin_amdgcn_global_load_async_to_lds_b128)
#define HAVE_ASYNC_LDS 1
// builtin signature (from hipcc diagnostic): params are pointers to
// gcc-vector int4 in AS1 (global) / AS3 (LDS), plus imm offset + cpol.
typedef int v4i_vs __attribute__((vector_size(16)));
typedef __attribute__((address_space(1))) v4i_vs* gv4i_p;
typedef __attribute__((address_space(3))) v4i_vs* lv4i_p;
static __device__ __forceinline__ void async_cp16(const float* g, float* l) {
  __builtin_amdgcn_global_load_async_to_lds_b128((gv4i_p)g, (lv4i_p)l, 0, 0);
}
#endif

static __device__ __forceinline__ v8f vzero8() {
  v8f z = {0.f,0.f,0.f,0.f,0.f,0.f,0.f,0.f};
  return z;
}
static __device__ __forceinline__ v2f ldf2(const float* p) {
  float2 t = *(const float2*)p;
  v2f r; r.x = t.x; r.y = t.y;
  return r;
}

// =====================================================================
// f32 WMMA GEMM: C[M x Nout] = A[M x K] @ W[K x Nout]
// One wave computes NT adjacent 16x16 output tiles (shared A operand).
// Requires Nout % (16*NT) == 0 and M % 16 == 0.  No predication.
// =====================================================================
template <int NT>
__global__ __launch_bounds__(32)
void gemm_f32_wmma(const float* __restrict__ A, const float* __restrict__ W,
                   float* __restrict__ C, int K, int Nout) {
  const int tileM = blockIdx.x, tileN = blockIdx.y;
  const int lane = threadIdx.x;
  const int half = lane >> 4;      // selects K pair (0,1) vs (2,3)
  const int l16  = lane & 15;
  const int row  = tileM * 16 + l16;
  const int col0 = tileN * (16 * NT) + l16;

  v8f acc[NT];
#pragma unroll
  for (int j = 0; j < NT; ++j) acc[j] = vzero8();

  const float* arow = A + (size_t)row * K;
  __builtin_prefetch(arow, 0, 3);
  __builtin_prefetch(W + col0, 0, 3);

#pragma unroll 4
  for (int k0 = 0; k0 < K; k0 += 4) {
    const int ka = k0 + half * 2;
    const v2f a = ldf2(arow + ka);
    const float* w0 = W + (size_t)ka * Nout + col0;
    const float* w1 = W + (size_t)(ka + 1) * Nout + col0;
#pragma unroll
    for (int j = 0; j < NT; ++j) {
      v2f b; b.x = w0[16 * j]; b.y = w1[16 * j];
      acc[j] = __builtin_amdgcn_wmma_f32_16x16x4_f32(false, a, false, b,
                                                     (short)0, acc[j], false, false);
    }
  }
#pragma unroll
  for (int j = 0; j < NT; ++j)
#pragma unroll
    for (int r = 0; r < 8; ++r) {
      const int m = tileM * 16 + r + half * 8;
      C[(size_t)m * Nout + col0 + 16 * j] = acc[j][r];
    }
}

// tiny projection for ww_idx (Nout = 2): plain dot products
__global__ __launch_bounds__(256)
void ww_proj(const float* __restrict__ A, const float* __restrict__ W,
             float* __restrict__ C) {
  const int idx = blockIdx.x * 256 + threadIdx.x;   // row * 2 + h
  const int row = idx >> 1, h = idx & 1;
  const float* arow = A + (size_t)row * D_;
  float s = 0.f;
#pragma unroll 8
  for (int k = 0; k < D_; ++k) s += arow[k] * W[k * IHEADS + h];
  C[idx] = s;
}

// =====================================================================
// Indexer scores for one 512-query chunk of one batch:
//   S[qloc, s] = sum_h widx[q,h] * relu( qi[q,h,:] . ki[s,:] )
// One wave per 16 queries x 32 keys (2 key tiles share the A operand).
// =====================================================================
__global__ __launch_bounds__(32)
void indexer_scores_wmma(const float* __restrict__ qi,   // (B,N,2,32)
                         const float* __restrict__ ki,   // (B,N,32)
                         const float* __restrict__ widx, // (B,N,2)
                         float* __restrict__ S,          // (CHUNK_Q, N)
                         int b, int q0) {
  const int qt = blockIdx.x, st = blockIdx.y;   // st covers 32 keys
  const int lane = threadIdx.x;
  const int half = lane >> 4;
  const int l16  = lane & 15;
  const int qrow = q0 + qt * 16 + l16;
  const int key0 = st * 32 + l16;
  const size_t qbase = ((size_t)b * N_ + qrow) * (IHEADS * IDIM);
  const size_t kb0 = ((size_t)b * N_ + key0) * IDIM;
  const size_t kb1 = kb0 + (size_t)16 * IDIM;

  float tot[2][8] = {{0.f,0.f,0.f,0.f,0.f,0.f,0.f,0.f},
                     {0.f,0.f,0.f,0.f,0.f,0.f,0.f,0.f}};
#pragma unroll
  for (int h = 0; h < IHEADS; ++h) {
    v8f a0 = vzero8(), a1 = vzero8();
    const float* arow = qi + qbase + h * IDIM;
#pragma unroll
    for (int k0 = 0; k0 < IDIM; k0 += 4) {
      const int ka = k0 + half * 2;
      const v2f a = ldf2(arow + ka);
      const v2f b0 = ldf2(ki + kb0 + ka);
      const v2f b1 = ldf2(ki + kb1 + ka);
      a0 = __builtin_amdgcn_wmma_f32_16x16x4_f32(false, a, false, b0,
                                                 (short)0, a0, false, false);
      a1 = __builtin_amdgcn_wmma_f32_16x16x4_f32(false, a, false, b1,
                                                 (short)0, a1, false, false);
    }
#pragma unroll
    for (int r = 0; r < 8; ++r) {
      const int qr = q0 + qt * 16 + r + half * 8;
      const float w = widx[((size_t)b * N_ + qr) * IHEADS + h];
      tot[0][r] += w * (a0[r] > 0.f ? a0[r] : 0.f);
      tot[1][r] += w * (a1[r] > 0.f ? a1[r] : 0.f);
    }
  }
#pragma unroll
  for (int j = 0; j < 2; ++j)
#pragma unroll
    for (int r = 0; r < 8; ++r) {
      const int qloc = qt * 16 + r + half * 8;
      S[(size_t)qloc * N_ + st * 32 + j * 16 + l16] = tot[j][r];
    }
}

// =====================================================================
// Top-128 of 4096 per query: 4-pass radix select over monotone keys.
// Selected SET equals the reference's (softmax+sum is order invariant).
// =====================================================================
static __device__ __forceinline__ unsigned f2key(float f) {
  unsigned u = __float_as_uint(f);
  return (u & 0x80000000u) ? ~u : (u | 0x80000000u);
}

__global__ __launch_bounds__(256)
void topk_radix(const float* __restrict__ S,   // (CHUNK_Q, N)
                int* __restrict__ top_idx,     // (B, N, TOPK)
                int b, int q0) {
  const int qloc = blockIdx.x;
  const float* row = S + (size_t)qloc * N_;
  int* out = top_idx + ((size_t)b * N_ + q0 + qloc) * TOPK;
  __builtin_prefetch(row + threadIdx.x, 0, 1);

  __shared__ unsigned hist[256];
  __shared__ unsigned sh_prefix;
  __shared__ int sh_rem, sh_gt, sh_eq;
  if (threadIdx.x == 0) { sh_prefix = 0u; sh_rem = TOPK; }
  __syncthreads();

  for (int pass = 0; pass < 4; ++pass) {
    const int shift = 24 - pass * 8;
    hist[threadIdx.x] = 0u;
    __syncthreads();
    const unsigned pfx = sh_prefix;
    for (int i = threadIdx.x; i < N_; i += 256) {
      const unsigned key = f2key(row[i]);
      const bool match = (pass == 0) || ((key >> (shift + 8)) == (pfx >> (shift + 8)));
      if (match) atomicAdd(&hist[(key >> shift) & 0xFFu], 1u);
    }
    __syncthreads();
    if (threadIdx.x == 0) {
      int rem = sh_rem;
      int acc = 0, chosen = 0;
      for (int bin = 255; bin >= 0; --bin) {
        const int h = (int)hist[bin];
        if (acc + h >= rem) { chosen = bin; sh_rem = rem - acc; break; }
        acc += h;
      }
      sh_prefix = pfx | ((unsigned)chosen << shift);
    }
    __syncthreads();
  }

  const unsigned T = sh_prefix;
  const int nties = sh_rem;
  const int G = TOPK - nties;
  if (threadIdx.x == 0) { sh_gt = 0; sh_eq = 0; }
  __syncthreads();
  for (int i = threadIdx.x; i < N_; i += 256) {
    const unsigned key = f2key(row[i]);
    if (key > T) {
      const int p = atomicAdd(&sh_gt, 1);
      if (p < G) out[p] = i;
    } else if (key == T) {
      const int p = atomicAdd(&sh_eq, 1);
      if (p < nties) out[G + p] = i;
    }
  }
}

// =====================================================================
// Sparse MQA over selected keys.  One wave per (b,q).
// K/V gather uses GLOBAL_LOAD_ASYNC_TO_LDS_B128 when available.
// =====================================================================
__global__ __launch_bounds__(32)
void sparse_attn_wmma(const float* __restrict__ qb,    // (B,N,8,48)
                      const float* __restrict__ kb,    // (B,N,48)
                      const float* __restrict__ vb,    // (B,N,48)
                      const int* __restrict__ top_idx, // (B,N,128)
                      float* __restrict__ ao) {        // (B,N,384)
  __shared__ float sk[TOPK * KDIM];   // 24 KB
  __shared__ float sv[TOPK * VDIM];   // 24 KB
  __shared__ float sp[16 * TOPK];     //  8 KB
  const int bq = blockIdx.x;          // b*N + q
  const int b  = bq >> 12;            // N_ == 4096
  const int lane = threadIdx.x;
  const int half = lane >> 4;
  const int l16  = lane & 15;

  // ---- gather 128 selected K/V rows (48 floats each) into LDS ----
  const int* sel = top_idx + (size_t)bq * TOPK;
#pragma unroll
  for (int jj = 0; jj < TOPK / 32; ++jj) {
    const int j = jj * 32 + lane;
    const int key = sel[j];
    const float* srck = kb + ((size_t)b * N_ + key) * KDIM;
    const float* srcv = vb + ((size_t)b * N_ + key) * VDIM;
#ifdef HAVE_ASYNC_LDS
#pragma unroll
    for (int t = 0; t < KDIM; t += 4) {
      async_cp16(srck + t, sk + j * KDIM + t);
      async_cp16(srcv + t, sv + j * VDIM + t);
    }
#else
    float4* dk = (float4*)(sk + j * KDIM);
    float4* dv = (float4*)(sv + j * VDIM);
#pragma unroll
    for (int t = 0; t < KDIM / 4; ++t) {
      dk[t] = ((const float4*)srck)[t];
      dv[t] = ((const float4*)srcv)[t];
    }
#endif
  }
#ifdef HAVE_ASYNC_LDS
#if __has_builtin(__builtin_amdgcn_s_wait_asynccnt)
  __builtin_amdgcn_s_wait_asynccnt(0);
#else
  asm volatile("s_wait_asynccnt 0x0" ::: "memory");
#endif
#endif
  __syncthreads();

  // ---- attn = Q(16x48, heads 0..7 valid) @ selK^T, 8 key tiles ----
  v8f acc[8];
#pragma unroll
  for (int t = 0; t < 8; ++t) acc[t] = vzero8();
  const float* qrow = qb + (size_t)bq * (NHEADS * KDIM);
#pragma unroll
  for (int k0 = 0; k0 < KDIM; k0 += 4) {
    const int ka = k0 + half * 2;
    v2f a;
    if (l16 < NHEADS) { a = ldf2(qrow + l16 * KDIM + ka); }
    else              { a.x = 0.f; a.y = 0.f; }
#pragma unroll
    for (int t = 0; t < 8; ++t) {
      const v2f bb = ldf2(sk + (t * 16 + l16) * KDIM + ka);   // ds_load_b64
      acc[t] = __builtin_amdgcn_wmma_f32_16x16x4_f32(false, a, false, bb,
                                                     (short)0, acc[t], false, false);
    }
  }

  // ---- softmax per head row (16-lane shfl reductions) ----
#pragma unroll
  for (int r = 0; r < 8; ++r) {
    float m = -3.0e38f;
#pragma unroll
    for (int t = 0; t < 8; ++t) { acc[t][r] *= SCALE_F; m = fmaxf(m, acc[t][r]); }
    for (int d = 1; d < 16; d <<= 1) m = fmaxf(m, __shfl_xor(m, d, 32));
    float s = 0.f;
#pragma unroll
    for (int t = 0; t < 8; ++t) { const float e = __expf(acc[t][r] - m); acc[t][r] = e; s += e; }
    for (int d = 1; d < 16; d <<= 1) s += __shfl_xor(s, d, 32);
    const float inv = 1.f / s;
#pragma unroll
    for (int t = 0; t < 8; ++t) acc[t][r] *= inv;
  }

  // ---- stage P (16 x 128 row-major) for A-layout reads ----
#pragma unroll
  for (int t = 0; t < 8; ++t)
#pragma unroll
    for (int r = 0; r < 8; ++r)
      sp[(r + half * 8) * TOPK + t * 16 + l16] = acc[t][r];
  __syncthreads();

  // ---- out = P(16x128) @ selV(128x48): shared A, 3 column tiles ----
  v8f o[3];
#pragma unroll
  for (int nt = 0; nt < 3; ++nt) o[nt] = vzero8();
#pragma unroll 4
  for (int k0 = 0; k0 < TOPK; k0 += 4) {
    const int ka = k0 + half * 2;
    const v2f a = ldf2(sp + l16 * TOPK + ka);                  // ds_load_b64
#pragma unroll
    for (int nt = 0; nt < 3; ++nt) {
      v2f bb;
      bb.x = sv[ka * VDIM + nt * 16 + l16];
      bb.y = sv[(ka + 1) * VDIM + nt * 16 + l16];
      o[nt] = __builtin_amdgcn_wmma_f32_16x16x4_f32(false, a, false, bb,
                                                    (short)0, o[nt], false, false);
    }
  }
  if (half == 0) {   // heads 0..7 valid only
#pragma unroll
    for (int nt = 0; nt < 3; ++nt)
#pragma unroll
      for (int r = 0; r < 8; ++r)
        ao[(size_t)bq * (NHEADS * VDIM) + r * VDIM + nt * 16 + l16] = o[nt][r];
  }
}

// =====================================================================
// launcher
// =====================================================================
extern "C" void kernel_launch(void* const* d_in, const int* in_sizes, int n_in,
                              void* d_out, int out_size, void* d_ws, size_t ws_size,
                              hipStream_t stream) {
  (void)in_sizes; (void)n_in; (void)out_size; (void)ws_size;
  const float* x      = (const float*)d_in[0];
  const float* wq_idx = (const float*)d_in[1];
  const float* wk_idx = (const float*)d_in[2];
  const float* ww_idx = (const float*)d_in[3];
  const float* wq     = (const float*)d_in[4];
  const float* wk     = (const float*)d_in[5];
  const float* wv     = (const float*)d_in[6];
  const float* wo     = (const float*)d_in[7];
  float* out = (float*)d_out;

  const size_t M = (size_t)B_ * N_;   // 8192 rows
  float* ws   = (float*)d_ws;
  float* qi   = ws;                   // M * 64
  float* ki   = qi   + M * 64;        // M * 32
  float* widx = ki   + M * 32;        // M * 2
  float* qb   = widx + M * 2;         // M * 384
  float* kb   = qb   + M * 384;       // M * 48
  float* vb   = kb   + M * 48;        // M * 48
  float* ao   = vb   + M * 48;        // M * 384
  int*   tidx = (int*)(ao + M * 384); // M * 128 ints
  float* sch  = (float*)(tidx + M * TOPK);  // CHUNK_Q * N floats

  const dim3 wv32(32);
  const int MT = (int)(M / 16);       // 512 row tiles

  // projections: Y = x @ W   (NT tuned so Nout % (16*NT) == 0)
  gemm_f32_wmma<2><<<dim3(MT,  2), wv32, 0, stream>>>(x, wq_idx, qi, D_, IHEADS * IDIM);
  gemm_f32_wmma<2><<<dim3(MT,  1), wv32, 0, stream>>>(x, wk_idx, ki, D_, IDIM);
  ww_proj<<<dim3((unsigned)(M * IHEADS / 256)), dim3(256), 0, stream>>>(x, ww_idx, widx);
  gemm_f32_wmma<4><<<dim3(MT,  6), wv32, 0, stream>>>(x, wq, qb, D_, NHEADS * KDIM);
  gemm_f32_wmma<3><<<dim3(MT,  1), wv32, 0, stream>>>(x, wk, kb, D_, KDIM);
  gemm_f32_wmma<3><<<dim3(MT,  1), wv32, 0, stream>>>(x, wv, vb, D_, VDIM);

  // indexer scores + top-k, chunked so the score buffer stays L2-resident
  for (int b = 0; b < B_; ++b) {
    for (int c = 0; c < N_ / CHUNK_Q; ++c) {
      const int q0 = c * CHUNK_Q;
      indexer_scores_wmma<<<dim3(CHUNK_Q / 16, N_ / 32), wv32, 0, stream>>>(
          qi, ki, widx, sch, b, q0);
      topk_radix<<<dim3(CHUNK_Q), dim3(256), 0, stream>>>(sch, tidx, b, q0);
    }
  }

  // sparse attention over selected keys
  sparse_attn_wmma<<<dim3((unsigned)M), wv32, 0, stream>>>(qb, kb, vb, tidx, ao);

  // final projection: out = ao @ wo
  gemm_f32_wmma<4><<<dim3(MT,  6), wv32, 0, stream>>>(ao, wo, out, NHEADS * VDIM, D_);
}